// HAN_46591805227167
// MI455X (gfx1250) — compile-verified
//
#include <hip/hip_runtime.h>
#include <math.h>

typedef float v2f __attribute__((ext_vector_type(2)));
typedef float v8f __attribute__((ext_vector_type(8)));

#define LEAKY(x) ((x) > 0.f ? (x) : 0.2f * (x))

// Native no-return f32 atomic add at device scope (avoids any CAS-loop expansion
// of atomicrmw fadd; lowers to exactly one GLOBAL_ATOMIC_ADD_F32 per call).
__device__ __forceinline__ void atomAddF(float* p, float v) {
  asm volatile("global_atomic_add_f32 %0, %1, off scope:SCOPE_DEV"
               :
               : "v"(p), "v"(v)
               : "memory");
}

// float atomic max via monotone integer punning (positive: int max, negative: uint min)
__device__ __forceinline__ void atomMaxF(float* p, float v) {
  if (v >= 0.f) atomicMax((int*)p, __float_as_int(v));
  else          atomicMin((unsigned int*)p, __float_as_uint(v));
}

// ---------------------------------------------------------------------------
// Y[N,64] = X[N,64] @ W[64,64] + b[64]   (N % 16 == 0)
// One wave computes a 16x16 output tile with 16x V_WMMA_F32_16X16X4_F32.
// Block = 128 threads (4 waves) covering one 16-row slab x all 64 columns.
// ---------------------------------------------------------------------------
__global__ void gemm64_bias(const float* __restrict__ X, const float* __restrict__ W,
                            const float* __restrict__ B, float* __restrict__ Y) {
  const int wave = threadIdx.x >> 5;
  const int lane = threadIdx.x & 31;
  const int half = lane >> 4;      // 0: lanes 0-15, 1: lanes 16-31
  const int l    = lane & 15;
  const int m0   = blockIdx.x << 4;
  const int col  = (wave << 4) + l;
  const int row  = m0 + l;

  v8f acc;
  const float bv = B[col];         // bias is per-column; each lane owns one column
#pragma unroll
  for (int j = 0; j < 8; ++j) acc[j] = bv;

#pragma unroll
  for (int k = 0; k < 16; ++k) {
    const int kk = (k << 2) + (half << 1);   // K pair for this half-wave
    v2f a, bm;
    a.x  = X[row * 64 + kk];       a.y  = X[row * 64 + kk + 1];
    bm.x = W[kk * 64 + col];       bm.y = W[(kk + 1) * 64 + col];
    acc = __builtin_amdgcn_wmma_f32_16x16x4_f32(false, a, false, bm,
                                                (short)0, acc, false, false);
  }
  // C/D layout: VGPR j -> row m0 + half*8 + j, column = col
#pragma unroll
  for (int j = 0; j < 8; ++j)
    Y[(size_t)(m0 + (half << 3) + j) * 64 + col] = acc[j];
}

// ---------------------------------------------------------------------------
// colsum[c] += sum_rows tanh( (X @ W + b)[row, c] )   (semantic attention stats)
// Same WMMA tile structure; each lane's 8 accumulators share a single column.
// ---------------------------------------------------------------------------
__global__ void gemm64_tanh_colsum(const float* __restrict__ X, const float* __restrict__ W,
                                   const float* __restrict__ B, float* __restrict__ colsum) {
  const int wave = threadIdx.x >> 5;
  const int lane = threadIdx.x & 31;
  const int half = lane >> 4;
  const int l    = lane & 15;
  const int m0   = blockIdx.x << 4;
  const int col  = (wave << 4) + l;
  const int row  = m0 + l;

  v8f acc;
  const float bv = B[col];
#pragma unroll
  for (int j = 0; j < 8; ++j) acc[j] = bv;

#pragma unroll
  for (int k = 0; k < 16; ++k) {
    const int kk = (k << 2) + (half << 1);
    v2f a, bm;
    a.x  = X[row * 64 + kk];       a.y  = X[row * 64 + kk + 1];
    bm.x = W[kk * 64 + col];       bm.y = W[(kk + 1) * 64 + col];
    acc = __builtin_amdgcn_wmma_f32_16x16x4_f32(false, a, false, bm,
                                                (short)0, acc, false, false);
  }
  float part = 0.f;
#pragma unroll
  for (int j = 0; j < 8; ++j) part += tanhf(acc[j]);
  atomAddF(&colsum[col], part);
}

// out[n,h] = sum_d h[n, h*32+d] * avec[h*32+d]
__global__ void attn_coef(const float* __restrict__ h, const float* __restrict__ avec,
                          float* __restrict__ out, int N) {
  int i = blockIdx.x * blockDim.x + threadIdx.x;   // over N*2
  if (i >= N * 2) return;
  const int n = i >> 1, hh = i & 1;
  const float* hp = h + (size_t)n * 64 + hh * 32;
  const float* ap = avec + hh * 32;
  float s = 0.f;
#pragma unroll
  for (int d = 0; d < 32; ++d) s += hp[d] * ap[d];
  out[i] = s;
}

__global__ void fillf(float* __restrict__ p, float v, int n) {
  int i = blockIdx.x * blockDim.x + threadIdx.x;
  if (i < n) p[i] = v;
}

// pass 1: segment max of leaky-relu logits
__global__ void edge_max(const int* __restrict__ src, const int* __restrict__ dst,
                         const float* __restrict__ as_, const float* __restrict__ ad_,
                         float* __restrict__ m, int E) {
  int e = blockIdx.x * blockDim.x + threadIdx.x;
  if (e >= E) return;
  const int s0 = src[e], d0 = dst[e];
#pragma unroll
  for (int hh = 0; hh < 2; ++hh) {
    float x = as_[s0 * 2 + hh] + ad_[d0 * 2 + hh];
    x = LEAKY(x);
    atomMaxF(&m[d0 * 2 + hh], x);
  }
}

// pass 2: exponentials + segment sum
__global__ void edge_exp(const int* __restrict__ src, const int* __restrict__ dst,
                         const float* __restrict__ as_, const float* __restrict__ ad_,
                         const float* __restrict__ m, float* __restrict__ ssum,
                         float* __restrict__ eb, int E) {
  int e = blockIdx.x * blockDim.x + threadIdx.x;
  if (e >= E) return;
  const int s0 = src[e], d0 = dst[e];
#pragma unroll
  for (int hh = 0; hh < 2; ++hh) {
    float x = as_[s0 * 2 + hh] + ad_[d0 * 2 + hh];
    x = LEAKY(x);
    const float ex = __expf(x - m[d0 * 2 + hh]);
    eb[e * 2 + hh] = ex;
    atomAddF(&ssum[d0 * 2 + hh], ex);
  }
}

// pass 3: agg[dst, c] += exp_e[head(c)] * h_src[src, c]   (division deferred)
__global__ void edge_scatter(const int* __restrict__ src, const int* __restrict__ dst,
                             const float* __restrict__ hsrc, const float* __restrict__ eb,
                             float* __restrict__ agg, int total) {
  int i = blockIdx.x * blockDim.x + threadIdx.x;   // over E*64
  if (i >= total) return;
  const int e = i >> 6, c = i & 63, hh = c >> 5;
  const int s0 = src[e], d0 = dst[e];
  atomAddF(&agg[(size_t)d0 * 64 + c], eb[e * 2 + hh] * hsrc[(size_t)s0 * 64 + c]);
}

// out = relu(agg / (s + 1e-16)), in place
__global__ void norm_relu(float* __restrict__ agg, const float* __restrict__ ssum, int total) {
  int i = blockIdx.x * blockDim.x + threadIdx.x;   // over N*64
  if (i >= total) return;
  const int n = i >> 6, hh = (i >> 5) & 1;
  float v = agg[i] / (ssum[n * 2 + hh] + 1e-16f);
  agg[i] = v > 0.f ? v : 0.f;
}

// scores s_k = sum_c q[c] * colsum[k,c] / N ; w = softmax(s)   (K = 2)
__global__ void scorew(const float* __restrict__ colsum, const float* __restrict__ q,
                       float invN, float* __restrict__ w) {
  __shared__ float red0[64];
  __shared__ float red1[64];
  const int c = threadIdx.x;   // 64 threads
  red0[c] = q[c] * colsum[c] * invN;
  red1[c] = q[c] * colsum[64 + c] * invN;
  __syncthreads();
  for (int st = 32; st > 0; st >>= 1) {
    if (c < st) { red0[c] += red0[c + st]; red1[c] += red1[c + st]; }
    __syncthreads();
  }
  if (c == 0) {
    const float s0 = red0[0], s1 = red1[0];
    const float mx = fmaxf(s0, s1);
    const float e0 = __expf(s0 - mx), e1 = __expf(s1 - mx);
    const float d = e0 + e1;
    w[0] = e0 / d; w[1] = e1 / d;
  }
}

__global__ void combine2(const float* __restrict__ o0, const float* __restrict__ o1,
                         const float* __restrict__ w, float* __restrict__ out, int total) {
  int i = blockIdx.x * blockDim.x + threadIdx.x;
  if (i >= total) return;
  out[i] = w[0] * o0[i] + w[1] * o1[i];
}

extern "C" void kernel_launch(void* const* d_in, const int* in_sizes, int n_in,
                              void* d_out, int out_size, void* d_ws, size_t ws_size,
                              hipStream_t stream) {
  (void)n_in; (void)out_size; (void)ws_size;
  const int* srcE[4] = {(const int*)d_in[3], (const int*)d_in[5],
                        (const int*)d_in[7], (const int*)d_in[9]};
  const int* dstE[4] = {(const int*)d_in[4], (const int*)d_in[6],
                        (const int*)d_in[8], (const int*)d_in[10]};
  const float* Wp   = (const float*)d_in[11];
  const float* bp   = (const float*)d_in[12];
  const float* aSrc = (const float*)d_in[13];
  const float* aDst = (const float*)d_in[14];
  const float* kW   = (const float*)d_in[15];
  const float* kb   = (const float*)d_in[16];
  const float* qv   = (const float*)d_in[17];
  const float* linW = (const float*)d_in[18];
  const float* linB = (const float*)d_in[19];

  const int NM = in_sizes[0] / 64, NS = in_sizes[1] / 64, NT = in_sizes[2] / 64;
  const int EMT = in_sizes[3], EST = in_sizes[7];
  const int Nn[3]   = {NM, NS, NT};
  const int Esz[4]  = {EMT, EMT, EST, EST};
  const int srcT[4] = {0, 2, 1, 2};      // member->team, team->member, skill->team, team->skill
  const int dstT[4] = {2, 0, 2, 1};

  // ---- workspace carve-up (floats) ----
  size_t off = 0;
  float* base = (float*)d_ws;
  auto take = [&](size_t n) { float* p = base + off; off += n; return p; };
  float* h[3];   for (int i = 0; i < 3; ++i) h[i] = take((size_t)Nn[i] * 64);
  float* aS[4];  float* aD[4];
  for (int e = 0; e < 4; ++e) { aS[e] = take((size_t)Nn[srcT[e]] * 2);
                                aD[e] = take((size_t)Nn[dstT[e]] * 2); }
  float* mB[4];  for (int e = 0; e < 4; ++e) mB[e]  = take((size_t)Nn[dstT[e]] * 2);
  float* sB[4];  for (int e = 0; e < 4; ++e) sB[e]  = take((size_t)Nn[dstT[e]] * 2);
  float* eb[4];  for (int e = 0; e < 4; ++e) eb[e]  = take((size_t)Esz[e] * 2);
  float* agg[4]; for (int e = 0; e < 4; ++e) agg[e] = take((size_t)Nn[dstT[e]] * 64);
  float* xTeam  = take((size_t)NT * 64);
  float* colsum = take(128);
  float* wsem   = take(2);

  const float* x[3] = {(const float*)d_in[0], (const float*)d_in[1], (const float*)d_in[2]};

  for (int L = 0; L < 3; ++L) {
    // 1) per-node-type projection h = x @ Wp + bp   (WMMA f32)
    for (int i = 0; i < 3; ++i)
      gemm64_bias<<<Nn[i] / 16, 128, 0, stream>>>(
          x[i], Wp + (size_t)(L * 3 + i) * 4096, bp + (L * 3 + i) * 64, h[i]);

    // 2) attention coefficients a_s, a_d per edge type
    for (int e = 0; e < 4; ++e) {
      attn_coef<<<(Nn[srcT[e]] * 2 + 255) / 256, 256, 0, stream>>>(
          h[srcT[e]], aSrc + (L * 4 + e) * 64, aS[e], Nn[srcT[e]]);
      attn_coef<<<(Nn[dstT[e]] * 2 + 255) / 256, 256, 0, stream>>>(
          h[dstT[e]], aDst + (L * 4 + e) * 64, aD[e], Nn[dstT[e]]);
    }

    // 3) init per-destination buffers
    for (int e = 0; e < 4; ++e) {
      const int n2 = Nn[dstT[e]] * 2, n64 = Nn[dstT[e]] * 64;
      fillf<<<(n2 + 255) / 256, 256, 0, stream>>>(mB[e], -3.402823466e38f, n2);
      fillf<<<(n2 + 255) / 256, 256, 0, stream>>>(sB[e], 0.f, n2);
      fillf<<<(n64 + 255) / 256, 256, 0, stream>>>(agg[e], 0.f, n64);
    }

    // 4) segment max, 5) exp + segment sum, 6) message scatter
    for (int e = 0; e < 4; ++e)
      edge_max<<<(Esz[e] + 255) / 256, 256, 0, stream>>>(srcE[e], dstE[e], aS[e], aD[e], mB[e], Esz[e]);
    for (int e = 0; e < 4; ++e)
      edge_exp<<<(Esz[e] + 255) / 256, 256, 0, stream>>>(srcE[e], dstE[e], aS[e], aD[e],
                                                         mB[e], sB[e], eb[e], Esz[e]);
    for (int e = 0; e < 4; ++e) {
      const int tot = Esz[e] * 64;
      edge_scatter<<<(tot + 255) / 256, 256, 0, stream>>>(srcE[e], dstE[e], h[srcT[e]],
                                                          eb[e], agg[e], tot);
    }

    // 7) normalize + relu (in place)
    for (int e = 0; e < 4; ++e) {
      const int tot = Nn[dstT[e]] * 64;
      norm_relu<<<(tot + 255) / 256, 256, 0, stream>>>(agg[e], sB[e], tot);
    }

    // 8) semantic attention — only team (K=2: edge types 0 and 2); K=1 types pass through
    fillf<<<1, 128, 0, stream>>>(colsum, 0.f, 128);
    gemm64_tanh_colsum<<<NT / 16, 128, 0, stream>>>(agg[0], kW + (size_t)L * 4096, kb + L * 64, colsum);
    gemm64_tanh_colsum<<<NT / 16, 128, 0, stream>>>(agg[2], kW + (size_t)L * 4096, kb + L * 64, colsum + 64);
    scorew<<<1, 64, 0, stream>>>(colsum, qv + L * 64, 1.0f / (float)NT, wsem);
    combine2<<<(NT * 64 + 255) / 256, 256, 0, stream>>>(agg[0], agg[2], wsem, xTeam, NT * 64);

    x[0] = agg[1]; x[1] = agg[3]; x[2] = xTeam;
  }

  // final shared Linear per type -> d_out (member | skill | team)
  float* out = (float*)d_out;
  gemm64_bias<<<NM / 16, 128, 0, stream>>>(x[0], linW, linB, out);
  gemm64_bias<<<NS / 16, 128, 0, stream>>>(x[1], linW, linB, out + (size_t)NM * 64);
  gemm64_bias<<<NT / 16, 128, 0, stream>>>(x[2], linW, linB, out + (size_t)(NM + NS) * 64);
}